// BSplineKanLayer_17222818857341
// MI455X (gfx1250) — compile-verified
//
#include <hip/hip_runtime.h>
#include <math.h>

#define IN_DIM   1024
#define OUT_DIM  1024
#define BATCH    2048
#define NBASES   13               // NUM_KNOTS - SPLINE_ORDER
#define BM       128
#define BN       128
#define ICH      4                // input dims per K-chunk
#define KC       (ICH * NBASES)   // 52 flattened-K per chunk (multiple of 4)
#define LDK      KC               // LDS row stride (dwords): 208B rows, 8B aligned,
                                  // 52*lane mod 64 is a permutation -> conflict-free
#define THREADS  256              // 8 waves (wave32)
#define A_FLOATS (BM * LDK)       // one A buffer (floats)
#define B_FLOATS (BN * LDK)       // one B buffer (floats)
#define SMEM_BYTES (4u * (2u * A_FLOATS + 2u * B_FLOATS))   // 106496 B

typedef __attribute__((ext_vector_type(2))) float        v2f;
typedef __attribute__((ext_vector_type(8))) float        v8f;
typedef __attribute__((ext_vector_type(4))) unsigned int v4u;
typedef __attribute__((ext_vector_type(4))) int          v4i;
typedef __attribute__((ext_vector_type(8))) int          v8i;

// Quadratic B-spline bases on uniform knots linspace(-1.125, 1.125, 16), h=0.15.
__device__ __forceinline__ void bspline_bases13(float t, float* __restrict__ o) {
  const float k0  = -1.125f;
  const float h   = 2.25f / 15.0f;
  const float ih  = 1.0f / h;
  const float i2h = 0.5f * ih;
  float b0[15];
#pragma unroll
  for (int j = 0; j < 15; ++j) {
    float kj = k0 + h * (float)j;
    b0[j] = (kj <= t && t < kj + h) ? 1.0f : 0.0f;
  }
  float b1[14];
#pragma unroll
  for (int j = 0; j < 14; ++j) {
    float kj = k0 + h * (float)j;
    b1[j] = (t - kj) * ih * b0[j] + ((kj + 2.0f * h) - t) * ih * b0[j + 1];
  }
#pragma unroll
  for (int j = 0; j < 13; ++j) {
    float kj = k0 + h * (float)j;
    o[j] = (t - kj) * i2h * b1[j] + ((kj + 3.0f * h) - t) * i2h * b1[j + 1];
  }
}

// TDM: DMA a 2D f32 tile (tile_h rows x tile_w elts, row stride in elts) from
// global memory into LDS as packed rows at lds_byte_addr. D# per ISA ch.8:
// group0 = {count=1 | lds_addr | global_addr[56:0] | type=2}
// group1 = {mask=0, data_size=4B | tensor_dim0/1 = tile dims (no OOB clamp) |
//           tile_dim0/1 | tensor_dim0_stride}
__device__ __forceinline__ void tdm_load_tile_f32(unsigned lds_byte_addr,
                                                  const float* gsrc,
                                                  unsigned tile_w, unsigned tile_h,
                                                  unsigned long long row_stride) {
  unsigned long long ga = (unsigned long long)(const void*)gsrc;
  v4u g0;
  g0.x = 1u;                                               // count=1 (valid user D#)
  g0.y = lds_byte_addr;                                    // D#.lds_addr
  g0.z = (unsigned)ga;                                     // global_addr[31:0]
  g0.w = (unsigned)((ga >> 32) & 0x01FFFFFFu) | (2u << 30);// ga[56:32] | type=2
  v8i g1;
  g1[0] = (int)(2u << 16);                                 // data_size=2 (4B)
  g1[1] = (int)((tile_w & 0xFFFFu) << 16);                 // tensor_dim0[15:0]
  g1[2] = (int)(((tile_w >> 16) & 0xFFFFu) |               // tensor_dim0[31:16]
                ((tile_h & 0xFFFFu) << 16));               // tensor_dim1[15:0]
  g1[3] = (int)(((tile_h >> 16) & 0xFFFFu) |               // tensor_dim1[31:16]
                ((tile_w & 0xFFFFu) << 16));               // tile_dim0
  g1[4] = (int)(tile_h & 0xFFFFu);                         // tile_dim1 (tile_dim2=0)
  g1[5] = (int)(row_stride & 0xFFFFFFFFull);               // dim0_stride[31:0]
  g1[6] = (int)((row_stride >> 32) & 0xFFFFull);           // dim0_stride[47:32]
  g1[7] = 0;
  v4i z4 = {0, 0, 0, 0};
#if defined(__clang_major__) && (__clang_major__ >= 23)
  v8i z8 = {0, 0, 0, 0, 0, 0, 0, 0};
  __builtin_amdgcn_tensor_load_to_lds(g0, g1, z4, z4, z8, 0);
#else
  __builtin_amdgcn_tensor_load_to_lds(g0, g1, z4, z4, 0);
#endif
}

// Compute spline bases for x[b0:b0+BM, ii:ii+ICH] -> As (K-major, stride LDK)
__device__ __forceinline__ void stage_bases(const float* __restrict__ x, int b0,
                                            int ii, int tid, float* __restrict__ As) {
  for (int p = tid; p < BM * ICH; p += THREADS) {
    const int bl = p >> 2, il = p & 3;
    const float t = x[(size_t)(b0 + bl) * IN_DIM + (ii + il)];
    float bas[NBASES];
    bspline_bases13(t, bas);
    float* dst = &As[bl * LDK + il * NBASES];
#pragma unroll
    for (int k = 0; k < NBASES; ++k) dst[k] = bas[k];
  }
}

// Kernel 1: per-row sum of silu(x) -> ws[b]
__global__ __launch_bounds__(THREADS)
void silu_rowsum_kernel(const float* __restrict__ x, float* __restrict__ out) {
  __shared__ float red[THREADS];
  const int b = blockIdx.x;
  const float* row = x + (size_t)b * IN_DIM;
  float s = 0.0f;
  for (int i = threadIdx.x; i < IN_DIM; i += THREADS) {
    float v = row[i];
    s += v / (1.0f + __expf(-v));
  }
  red[threadIdx.x] = s;
  __syncthreads();
  for (int off = THREADS / 2; off > 0; off >>= 1) {
    if (threadIdx.x < off) red[threadIdx.x] += red[threadIdx.x + off];
    __syncthreads();
  }
  if (threadIdx.x == 0) out[b] = red[0];
}

// Kernel 2: fused bases + einsum('bik,oik->bo') via V_WMMA_F32_16X16X4_F32,
// TDM-staged params, double-buffered LDS, epilogue out = w_b*silu + w_s*splines.
__global__ __launch_bounds__(THREADS)
void kan_spline_wmma_kernel(const float* __restrict__ x,
                            const float* __restrict__ P,      // [OUT][IN][13]
                            const float* __restrict__ wb_p,
                            const float* __restrict__ ws_p,
                            const float* __restrict__ silu_sum,
                            float* __restrict__ out) {
  extern __shared__ float smem[];
  float* const AsBase = smem;                       // 2 x A_FLOATS
  float* const BsBase = smem + 2 * A_FLOATS;        // 2 x B_FLOATS
  // Low 32 bits of a generic LDS pointer == wave-relative LDS byte address.
  const unsigned bs_lds_base = (unsigned)(uintptr_t)(void*)BsBase;

  const int tid  = threadIdx.x;
  const int lane = tid & 31;      // wave32
  const int wave = tid >> 5;      // 0..7
  const int wm   = wave >> 2;     // 0..1 : 64-row slice
  const int wn   = wave & 3;      // 0..3 : 32-col slice
  const int lo   = lane & 15;
  const int hi   = lane >> 4;     // K pair {0,1} vs {2,3}

  const int b0 = blockIdx.y * BM;
  const int o0 = blockIdx.x * BN;
  const float* Prow = P + (size_t)o0 * IN_DIM * NBASES;

  v8f acc[4][2];
#pragma unroll
  for (int mt = 0; mt < 4; ++mt)
#pragma unroll
    for (int nt = 0; nt < 2; ++nt)
      acc[mt][nt] = {};

  // ---- prologue: fill buffer 0 ----
  if (wave == 0)
    tdm_load_tile_f32(bs_lds_base, Prow, KC, BN,
                      (unsigned long long)IN_DIM * NBASES);
  stage_bases(x, b0, 0, tid, AsBase);
  if (wave == 0) __builtin_amdgcn_s_wait_tensorcnt(0);
  __syncthreads();

  int cur = 0;
  for (int ii = 0; ii < IN_DIM; ii += ICH) {
    const int nxt = cur ^ 1;
    const bool have_next = (ii + ICH) < IN_DIM;

    // ---- prefetch next chunk while this one computes ----
    if (have_next) {
      if (wave == 0)
        tdm_load_tile_f32(bs_lds_base + (unsigned)nxt * (unsigned)(B_FLOATS * 4),
                          Prow + (size_t)(ii + ICH) * NBASES,
                          KC, BN, (unsigned long long)IN_DIM * NBASES);
      stage_bases(x, b0, ii + ICH, tid, AsBase + nxt * A_FLOATS);
    }

    // ---- 13 K4 steps of f32 WMMA over the flattened 52-wide K chunk ----
    const float* As = AsBase + cur * A_FLOATS;
    const float* Bs = BsBase + cur * B_FLOATS;
#pragma unroll
    for (int ks = 0; ks < KC / 4; ++ks) {
      const int col = ks * 4 + 2 * hi;  // lanes 0-15 hold {K0,K1}, 16-31 {K2,K3}
      v2f a[4], bf[2];
#pragma unroll
      for (int mt = 0; mt < 4; ++mt)
        a[mt] = *(const v2f*)&As[(wm * 64 + mt * 16 + lo) * LDK + col];
#pragma unroll
      for (int nt = 0; nt < 2; ++nt)
        bf[nt] = *(const v2f*)&Bs[(wn * 32 + nt * 16 + lo) * LDK + col];
#pragma unroll
      for (int mt = 0; mt < 4; ++mt)
#pragma unroll
        for (int nt = 0; nt < 2; ++nt)
          acc[mt][nt] = __builtin_amdgcn_wmma_f32_16x16x4_f32(
              false, a[mt], false, bf[nt], (short)0, acc[mt][nt], false, false);
    }

    if (wave == 0 && have_next) __builtin_amdgcn_s_wait_tensorcnt(0);
    __syncthreads();
    cur = nxt;
  }

  // ---- epilogue: out = w_b * silu_sum[b] + w_s * acc ----
  const float wb  = *wb_p;
  const float wsc = *ws_p;
#pragma unroll
  for (int mt = 0; mt < 4; ++mt) {
    const int rbase = b0 + wm * 64 + mt * 16 + 8 * hi;  // C layout: vgpr r -> M=r (+8 hi half)
#pragma unroll
    for (int nt = 0; nt < 2; ++nt) {
      const int col = o0 + wn * 32 + nt * 16 + lo;
#pragma unroll
      for (int r = 0; r < 8; ++r) {
        const int row = rbase + r;
        out[(size_t)row * OUT_DIM + col] = wb * silu_sum[row] + wsc * acc[mt][nt][r];
      }
    }
  }
}

extern "C" void kernel_launch(void* const* d_in, const int* in_sizes, int n_in,
                              void* d_out, int out_size, void* d_ws, size_t ws_size,
                              hipStream_t stream) {
  (void)in_sizes; (void)n_in; (void)out_size; (void)ws_size;
  const float* x   = (const float*)d_in[0];
  const float* P   = (const float*)d_in[1];
  const float* wb  = (const float*)d_in[2];
  const float* wsc = (const float*)d_in[3];
  float* out  = (float*)d_out;
  float* silu = (float*)d_ws;   // 2048 floats of scratch

  silu_rowsum_kernel<<<BATCH, THREADS, 0, stream>>>(x, silu);

  dim3 grid(OUT_DIM / BN, BATCH / BM);
  kan_spline_wmma_kernel<<<grid, THREADS, SMEM_BYTES, stream>>>(x, P, wb, wsc, silu, out);
}